// ReconstructLoss_35519379538319
// MI455X (gfx1250) — compile-verified
//
#include <hip/hip_runtime.h>
#include <math.h>

typedef __attribute__((ext_vector_type(2))) float v2f;
typedef __attribute__((ext_vector_type(8))) float v8f;

#define BATCH 4
#define NPTS  4096
#define EPS_V 1e-4f
#define LOG2E 1.4426950408889634f
#define LN2   0.6931471805599453f
// WMMA produces (x.y) * LOG2E/EPS directly: points pre-scaled by sqrt(LOG2E/EPS)
#define PSCALE   120.1122409f            /* sqrt(1e4 * LOG2E) */
#define TSCALE   14426.950408889634f     /* 1e4 * LOG2E        */
#define EPS_LN2  (EPS_V * LN2)           /* converts base-2 lse back to nats */
// -EPS * logw = EPS * log(4096)
#define NEG_EPS_LOGW (EPS_V * 8.317766166719343f)

// v_exp_f32 / v_log_f32 are base-2 in hardware; use the amdgcn builtins directly.
__device__ __forceinline__ float fast_exp2(float x) { return __builtin_amdgcn_exp2f(x); }
__device__ __forceinline__ float fast_log2(float x) { return __builtin_amdgcn_logf(x); }

// ---------------------------------------------------------------------------
// Prep: pack points into float4 scaled by sqrt(LOG2E/EPS) so the WMMA dot
// product is already in base-2 logsumexp units; compute 0.5|p|^2; zero the
// potential buffers.
// ---------------------------------------------------------------------------
__global__ __launch_bounds__(256) void sinkhorn_prep(
    const float* __restrict__ pq, const float* __restrict__ pgt,
    float4* __restrict__ Xs, float4* __restrict__ Ys,
    float* __restrict__ sqx, float* __restrict__ sqy,
    float* __restrict__ F, float* __restrict__ G,
    float* __restrict__ FA, float* __restrict__ GA)
{
    int i = blockIdx.x * 256 + threadIdx.x;   // [0, BATCH*NPTS)
    float x0 = pq[i*3+0], x1 = pq[i*3+1], x2 = pq[i*3+2];
    float y0 = pgt[i*4+0], y1 = pgt[i*4+1], y2 = pgt[i*4+2];
    Xs[i] = make_float4(x0*PSCALE, x1*PSCALE, x2*PSCALE, 0.0f);
    Ys[i] = make_float4(y0*PSCALE, y1*PSCALE, y2*PSCALE, 0.0f);
    sqx[i] = 0.5f*(x0*x0 + x1*x1 + x2*x2);
    sqy[i] = 0.5f*(y0*y0 + y1*y1 + y2*y2);
    F[i] = 0.0f; G[i] = 0.0f; FA[i] = 0.0f; GA[i] = 0.0f;
}

// ---------------------------------------------------------------------------
// One softmin pass, all-base-2 internally:
//   u[n,m] = LOG2E*( (inPot[m] - 0.5|y_m|^2 + x_n.y_m) / EPS )
//   out[b,n] = -EPS*logw + 0.5|x_n|^2 - EPS*LN2 * log2sumexp2_m( u[n,m] )
// t[m] = (inPot[m]-0.5|y_m|^2)*LOG2E/EPS staged in LDS and injected as the
// WMMA C accumulator; A (16x4) = reduce side (m), B (4x16) = output side (n).
// D layout: lane owns one output n (N=lane&15) and a private stream of m
// values (VGPR r -> M=r+8*(lane>>4)); online log2sumexp is lane-local with a
// single shfl_xor(16) merge. mixFlag: out = 0.5*(selfPot + out).
// ---------------------------------------------------------------------------
__global__ __launch_bounds__(256) void sinkhorn_softmin(
    float* __restrict__ outPot,
    const float4* __restrict__ Pout, const float* __restrict__ sqOut,
    const float4* __restrict__ Pred, const float* __restrict__ sqRed,
    const float* __restrict__ inPot, const float* __restrict__ selfPot,
    int mixFlag)
{
    __shared__ __align__(16) float Tsh[NPTS];
    const int b    = blockIdx.x >> 5;     // 32 blocks per batch (128 n per block)
    const int nblk = blockIdx.x & 31;
    const int tid  = threadIdx.x;

    for (int m = tid; m < NPTS; m += 256)
        Tsh[m] = (inPot[b*NPTS + m] - sqRed[b*NPTS + m]) * TSCALE;
    __syncthreads();

    const int lane = tid & 31;
    const int wid  = tid >> 5;
    const int half = lane >> 4;
    const int l15  = lane & 15;
    const int n    = nblk*128 + wid*16 + l15;

    // B matrix (4x16, K x N): VGPR r holds K=r (lanes 0-15) / K=r+2 (lanes 16-31)
    float4 xp = Pout[b*NPTS + n];
    v2f Bv;
    Bv.x = half ? xp.z : xp.x;
    Bv.y = half ? xp.w : xp.y;

    float rm = -INFINITY, ssum = 0.0f;
    const float4* PredB = Pred + b*NPTS;

    for (int mb = 0; mb < NPTS; mb += 16) {
        // A matrix (16x4): lane m-row = l15; VGPR0/1 = K0/K1 (lo) or K2/K3 (hi)
        float4 yp = PredB[mb + l15];
        v2f Av;
        Av.x = half ? yp.z : yp.x;
        Av.y = half ? yp.w : yp.y;

        // C accumulator: c[r] = t[mb + 8*half + r]
        const float4* tp = (const float4*)(&Tsh[mb + half*8]);
        float4 t0 = tp[0];
        float4 t1 = tp[1];
        v8f c;
        c[0]=t0.x; c[1]=t0.y; c[2]=t0.z; c[3]=t0.w;
        c[4]=t1.x; c[5]=t1.y; c[6]=t1.z; c[7]=t1.w;

        v8f d = __builtin_amdgcn_wmma_f32_16x16x4_f32(
            /*neg_a=*/false, Av, /*neg_b=*/false, Bv,
            /*c_mod=*/(short)0, c, /*reuse_a=*/false, /*reuse_b=*/false);

        // online log2-sum-exp2 over the 8 private m-values (pure v_exp_f32)
        float m8 = fmaxf(fmaxf(fmaxf(d[0],d[1]), fmaxf(d[2],d[3])),
                         fmaxf(fmaxf(d[4],d[5]), fmaxf(d[6],d[7])));
        float nm = fmaxf(rm, m8);
        float e = fast_exp2(d[0]-nm) + fast_exp2(d[1]-nm) + fast_exp2(d[2]-nm) + fast_exp2(d[3]-nm)
                + fast_exp2(d[4]-nm) + fast_exp2(d[5]-nm) + fast_exp2(d[6]-nm) + fast_exp2(d[7]-nm);
        ssum = ssum * fast_exp2(rm - nm) + e;
        rm = nm;
    }

    // merge the two m-half streams (lane <-> lane^16 share the same n)
    float rm2 = __shfl_xor(rm, 16, 32);
    float s2  = __shfl_xor(ssum, 16, 32);
    float nm  = fmaxf(rm, rm2);
    float st  = ssum * fast_exp2(rm - nm) + s2 * fast_exp2(rm2 - nm);
    float lse2 = nm + fast_log2(st);               // base-2 logsumexp

    float val = NEG_EPS_LOGW + sqOut[b*NPTS + n] - EPS_LN2 * lse2;
    if (mixFlag) val = 0.5f * (selfPot[b*NPTS + n] + val);
    if (half == 0) outPot[b*NPTS + n] = val;
}

// ---------------------------------------------------------------------------
// L[b] = mean_n(f_ab - f_aa) + mean_m(g_ab - g_bb)   (N == M == 4096)
// ---------------------------------------------------------------------------
__global__ __launch_bounds__(256) void sinkhorn_finalize(
    const float* __restrict__ F, const float* __restrict__ FAA,
    const float* __restrict__ G, const float* __restrict__ GBB,
    float* __restrict__ out)
{
    __shared__ float red[256];
    const int b = blockIdx.x;
    const int tid = threadIdx.x;
    float acc = 0.0f;
    for (int i = tid; i < NPTS; i += 256)
        acc += (F[b*NPTS+i] - FAA[b*NPTS+i]) + (G[b*NPTS+i] - GBB[b*NPTS+i]);
    red[tid] = acc;
    __syncthreads();
    for (int s = 128; s > 0; s >>= 1) {
        if (tid < s) red[tid] += red[tid + s];
        __syncthreads();
    }
    if (tid == 0) out[b] = red[0] * (1.0f / NPTS);
}

// ---------------------------------------------------------------------------
extern "C" void kernel_launch(void* const* d_in, const int* in_sizes, int n_in,
                              void* d_out, int out_size, void* d_ws, size_t ws_size,
                              hipStream_t stream)
{
    const float* pq  = (const float*)d_in[0];
    const float* pgt = (const float*)d_in[1];

    const int BN = BATCH * NPTS;                 // 16384
    float*  w   = (float*)d_ws;                  // ~1 MiB total
    float4* Xs  = (float4*)w;                    // BN float4
    float4* Ys  = Xs + BN;                       // BN float4
    float*  sqx = (float*)(Ys + BN);             // BN
    float*  sqy = sqx + BN;
    float*  F   = sqy + BN;                      // f_ab potential
    float*  G   = F + BN;                        // g_ab potential
    float*  FA  = G + BN;                        // f_aa ping
    float*  FB  = FA + BN;                       // f_aa pong
    float*  GA  = FB + BN;                       // g_bb ping
    float*  GB  = GA + BN;                       // g_bb pong

    sinkhorn_prep<<<BN/256, 256, 0, stream>>>(pq, pgt, Xs, Ys, sqx, sqy, F, G, FA, GA);

    dim3 smGrid(BATCH * (NPTS/128));

    // ab Sinkhorn loop: f = sm2(g) [reduce over y/m], g = sm1(f) [reduce over x/n]
    for (int it = 0; it < 20; ++it) {
        sinkhorn_softmin<<<smGrid, 256, 0, stream>>>(F, Xs, sqx, Ys, sqy, G, G, 0);
        sinkhorn_softmin<<<smGrid, 256, 0, stream>>>(G, Ys, sqy, Xs, sqx, F, F, 0);
    }
    // f_ab = sm2(g20); g_ab = sm1(f_ab)   (note: potentials' final g == g20)
    sinkhorn_softmin<<<smGrid, 256, 0, stream>>>(F, Xs, sqx, Ys, sqy, G, G, 0);
    sinkhorn_softmin<<<smGrid, 256, 0, stream>>>(G, Ys, sqy, Xs, sqx, F, F, 0);

    // symmetric potential xx: f <- 0.5*(f + sm2_xx(f)), then f_aa = sm2_xx(f20)
    {
        float* ca = FA; float* cb = FB;
        for (int it = 0; it < 20; ++it) {
            sinkhorn_softmin<<<smGrid, 256, 0, stream>>>(cb, Xs, sqx, Xs, sqx, ca, ca, 1);
            float* t = ca; ca = cb; cb = t;
        }
        // after 20 swaps: ca == FA (current), cb == FB; final (no mix) -> FB
        sinkhorn_softmin<<<smGrid, 256, 0, stream>>>(cb, Xs, sqx, Xs, sqx, ca, ca, 0);
    }
    // symmetric potential yy
    {
        float* ca = GA; float* cb = GB;
        for (int it = 0; it < 20; ++it) {
            sinkhorn_softmin<<<smGrid, 256, 0, stream>>>(cb, Ys, sqy, Ys, sqy, ca, ca, 1);
            float* t = ca; ca = cb; cb = t;
        }
        sinkhorn_softmin<<<smGrid, 256, 0, stream>>>(cb, Ys, sqy, Ys, sqy, ca, ca, 0);
    }

    sinkhorn_finalize<<<BATCH, 256, 0, stream>>>(F, FB, G, GB, (float*)d_out);
}